// GenomeWideHistogramMetric_16372415332487
// MI455X (gfx1250) — compile-verified
//
#include <hip/hip_runtime.h>
#include <hip/hip_bf16.h>
#include <cstdint>

#define NUM_BINS   1000
#define T_TRACKS   32
#define B_DIM      8
#define L_DIM      8192
#define SMOOTH     5
#define L_OUT      (L_DIM - SMOOTH + 1)          // 8188
#define TILE_IN    32                            // staged input rows per tile
#define TILE_OUT   (TILE_IN - SMOOTH + 1)        // 28 output rows per tile
#define TILES_PER_B ((L_OUT + TILE_OUT - 1) / TILE_OUT)   // 293
#define NTILES     (B_DIM * TILES_PER_B)         // 2344
#define WAVES_PER_BLOCK 8
#define BLOCK_THREADS   (WAVES_PER_BLOCK * 32)
#define ROW_BYTES  (T_TRACKS * 4)                // 128 B per (b,l,:) row
#define TILE_FLOATS (TILE_IN * T_TRACKS)         // 1024 floats = 4 KB

// ---------------------------------------------------------------------------
// Zero the 128 MB output (harness poisons it with 0xAA). float4 stores.
// ---------------------------------------------------------------------------
__global__ void hist_zero_kernel(float* __restrict__ out, long n) {
    long n4 = n >> 2;
    long i  = (long)blockIdx.x * blockDim.x + threadIdx.x;
    long st = (long)gridDim.x * blockDim.x;
    float4 z = make_float4(0.f, 0.f, 0.f, 0.f);
    float4* p4 = (float4*)out;
    for (long k = i; k < n4; k += st) p4[k] = z;
    // scalar tail
    for (long k = (n4 << 2) + i; k < n; k += st) out[k] = 0.f;
}

// Low 32 bits of a flat pointer into LDS == byte offset within the workgroup's
// LDS allocation (flat LDS aperture keeps the offset in addr[31:0]).
__device__ __forceinline__ unsigned lds_off32(const void* p) {
    return (unsigned)(unsigned long long)p;
}

// Stage TILE_IN rows x 32 floats (4 KB) of one tensor into LDS with the CDNA5
// async DMA path: one GLOBAL_LOAD_ASYNC_TO_LDS_B128 moves 32 lanes x 16 B =
// 512 B = 4 rows. 8 issues per tile; tracked on ASYNCcnt.
__device__ __forceinline__ void async_stage_tile(const float* __restrict__ gsrc_b,
                                                 int l0, const float* lds_tile,
                                                 int lane) {
    const int sub_row = lane >> 3;        // 0..3 : row within a 4-row group
    const int colb    = (lane & 7) * 16;  // byte column within the 128 B row
    unsigned lds_base = lds_off32(lds_tile) + (unsigned)(sub_row * ROW_BYTES + colb);
    #pragma unroll
    for (int g = 0; g < TILE_IN / 4; ++g) {
        int r = l0 + g * 4 + sub_row;                 // input row within batch b
        r = (r < L_DIM - 1) ? r : (L_DIM - 1);        // clamp halo at batch end
        const char* gaddr = (const char*)gsrc_b + (size_t)r * ROW_BYTES + colb;
        unsigned    laddr = lds_base + (unsigned)(g * 4 * ROW_BYTES);
        asm volatile("global_load_async_to_lds_b128 %0, %1, off"
                     :: "v"(laddr), "v"(gaddr) : "memory");
    }
}

// ---------------------------------------------------------------------------
// Histogram kernel: one wave per tile (grid-strided), double-buffered async
// LDS staging, register sliding window for the 5-tap mean, float atomics.
// ---------------------------------------------------------------------------
__global__ void __launch_bounds__(BLOCK_THREADS)
hist_kernel(const float* __restrict__ yp, const float* __restrict__ yt,
            float* __restrict__ out) {
    __shared__ float smem[WAVES_PER_BLOCK][2][2][TILE_FLOATS];  // 128 KB / WG

    float* __restrict__ pred_hist = out;
    float* __restrict__ tgt_hist  = out + (size_t)T_TRACKS * NUM_BINS;
    float* __restrict__ joint     = out + (size_t)2 * T_TRACKS * NUM_BINS;

    const int lane  = threadIdx.x & 31;          // lane == track id (wave32)
    const int wslot = threadIdx.x >> 5;
    const int wave  = (int)((blockIdx.x * BLOCK_THREADS + threadIdx.x) >> 5);
    const int nw    = (int)((gridDim.x * BLOCK_THREADS) >> 5);

    int tile = wave;
    if (tile >= NTILES) return;

    auto stage = [&](int t, int buf) {
        int b  = t / TILES_PER_B;
        int l0 = (t % TILES_PER_B) * TILE_OUT;
        const float* pb_ = yp + (size_t)b * L_DIM * T_TRACKS;
        const float* tb_ = yt + (size_t)b * L_DIM * T_TRACKS;
        async_stage_tile(pb_, l0, smem[wslot][buf][0], lane);
        async_stage_tile(tb_, l0, smem[wslot][buf][1], lane);
    };

    stage(tile, 0);                               // prologue DMA
    int buf = 0;
    for (; tile < NTILES; tile += nw, buf ^= 1) {
        // Current buffer's DMA done; prior DS reads retired before reuse.
        asm volatile("s_wait_asynccnt 0\n\ts_wait_dscnt 0" ::: "memory");

        int next = tile + nw;
        if (next < NTILES) stage(next, buf ^ 1);  // prefetch overlaps compute

        const float* __restrict__ sp = smem[wslot][buf][0];
        const float* __restrict__ st = smem[wslot][buf][1];
        int l0   = (tile % TILES_PER_B) * TILE_OUT;
        int nout = L_OUT - l0; nout = (nout > TILE_OUT) ? TILE_OUT : nout;

        // 5-tap sliding window in registers; LDS column reads are
        // conflict-free (bank = lane + 32*row mod 64).
        float pw[SMOOTH], tw[SMOOTH];
        #pragma unroll
        for (int i = 0; i < SMOOTH - 1; ++i) {
            pw[i] = sp[i * T_TRACKS + lane];
            tw[i] = st[i * T_TRACKS + lane];
        }
        for (int r = 0; r < nout; ++r) {
            pw[SMOOTH - 1] = sp[(r + SMOOTH - 1) * T_TRACKS + lane];
            tw[SMOOTH - 1] = st[(r + SMOOTH - 1) * T_TRACKS + lane];
            float vp = (pw[0] + pw[1] + pw[2] + pw[3] + pw[4]) / 5.0f;
            float vt = (tw[0] + tw[1] + tw[2] + tw[3] + tw[4]) / 5.0f;
            if (__builtin_isfinite(vp) && __builtin_isfinite(vt)) {
                float cp = fminf(fmaxf(vp, 0.0f), 10.0f);
                float ct = fminf(fmaxf(vt, 0.0f), 10.0f);
                int pbin = (int)(cp * 100.0f); pbin = pbin > NUM_BINS - 1 ? NUM_BINS - 1 : pbin;
                int tbin = (int)(ct * 100.0f); tbin = tbin > NUM_BINS - 1 ? NUM_BINS - 1 : tbin;
                // lane==track => no intra-wave address collisions on 1-D hists
                atomicAdd(&pred_hist[lane * NUM_BINS + pbin], 1.0f);
                atomicAdd(&tgt_hist [lane * NUM_BINS + tbin], 1.0f);
                atomicAdd(&joint[((size_t)lane * NUM_BINS + pbin) * NUM_BINS + tbin], 1.0f);
            }
            #pragma unroll
            for (int i = 0; i < SMOOTH - 1; ++i) { pw[i] = pw[i + 1]; tw[i] = tw[i + 1]; }
        }
    }
}

extern "C" void kernel_launch(void* const* d_in, const int* in_sizes, int n_in,
                              void* d_out, int out_size, void* d_ws, size_t ws_size,
                              hipStream_t stream) {
    const float* yp = (const float*)d_in[0];
    const float* yt = (const float*)d_in[1];
    float* out = (float*)d_out;

    // Pass 1: clear 32,064,000 output floats (~128 MB) — bandwidth-bound.
    hist_zero_kernel<<<4096, 256, 0, stream>>>(out, (long)out_size);

    // Pass 2: 2344 wave-tiles over 1024 waves -> each wave pipelines 2-3
    // tiles through the double-buffered async-LDS stage.
    hist_kernel<<<128, BLOCK_THREADS, 0, stream>>>(yp, yt, out);
}